// MainAttention_22385369547246
// MI455X (gfx1250) — compile-verified
//
#include <hip/hip_runtime.h>
#include <hip/hip_bf16.h>
#include <math.h>

// ---------------------------------------------------------------------------
// GLA block for MI455X (gfx1250, wave32).
//   B=2, T=2048, HID=1024, H=16, HKV=4, D=64, GLR=16
// Pipeline:
//   1) Q = relu(h@Wq), K = relu(h@Wk), V = h@Wv, Tmp = h@Wg1   [fp32 WMMA GEMMs]
//   2) G = logsigmoid(Tmp@Wg2 + bg2) / 16                       [VALU]
//   3) GLA scan, TDM-staged LDS tiles                           [register state]
//   4) per-head RMSNorm over D=64                               [wave32 shuffles]
//   5) out = O @ Wo                                             [fp32 WMMA GEMM]
// ---------------------------------------------------------------------------

typedef __attribute__((ext_vector_type(2))) float    v2f;
typedef __attribute__((ext_vector_type(8))) float    v8f;
typedef __attribute__((ext_vector_type(4))) unsigned u32x4;
typedef __attribute__((ext_vector_type(8))) unsigned u32x8;

#define GLA_B    2
#define GLA_T    2048
#define GLA_HID  1024
#define GLA_H    16
#define GLA_HKV  4
#define GLA_D    64
#define GLA_TOK  (GLA_B * GLA_T)          // 4096
#define GLA_KVW  (GLA_HKV * GLA_D)        // 256

__device__ __forceinline__ v8f wmma_f32_k4(v2f a, v2f b, v8f c) {
  // V_WMMA_F32_16X16X4_F32 : D = A(16x4,f32) * B(4x16,f32) + C(16x16,f32)
  return __builtin_amdgcn_wmma_f32_16x16x4_f32(
      /*neg_a=*/false, a, /*neg_b=*/false, b,
      /*c_mod=*/(short)0, c, /*reuse_a=*/false, /*reuse_b=*/false);
}

// ---------------------------------------------------------------------------
// C[M,N] = (relu?) A[M,K] @ B[K,N].  One wave computes a 16 x (16*NT) strip:
// one A fragment is reused across NT B fragments per K-step, so VMEM
// instructions per WMMA drop from 3 to ~2.25 and A stays in registers.
// Fragment layouts per ISA 7.12.2 (32-bit A 16x4 / mirrored B / 8-VGPR C).
// ---------------------------------------------------------------------------
template <int NT>
__global__ void __launch_bounds__(256)
gemm_f32_wmma(const float* __restrict__ A, const float* __restrict__ B,
              float* __restrict__ C, int M, int N, int K, int do_relu)
{
  const int lane = threadIdx.x & 31;
  const int wave = threadIdx.x >> 5;
  const int nblk = N / (16 * NT);
  const int tile = blockIdx.x * 8 + wave;
  if (tile >= (M >> 4) * nblk) return;

  const int m0  = (tile / nblk) << 4;
  const int n0  = (tile % nblk) * (16 * NT);
  const int row = lane & 15;              // M row (A) / N col (B)
  const int kh  = (lane >> 4) << 1;       // 0 or 2 : K sub-offset for this lane

  const float* Ap = A + (size_t)(m0 + row) * K + kh;
  const float* Bp = B + (size_t)kh * N + (n0 + row);

  v8f acc[NT];
  #pragma unroll
  for (int nt = 0; nt < NT; ++nt) acc[nt] = (v8f){};

  for (int kk = 0; kk < K; kk += 32) {
    // pull the next K-chunk toward the WGP while this one computes
    __builtin_prefetch(Ap + kk + 128, 0, 0);
    __builtin_prefetch(Bp + (size_t)(kk + 32) * N, 0, 0);
    #pragma unroll
    for (int u = 0; u < 8; ++u) {
      const int k0 = kk + u * 4;
      const float2 av = *(const float2*)(Ap + k0);   // A[m, k0+kh .. +1]
      v2f a;
      a.x = av.x;
      a.y = av.y;
      #pragma unroll
      for (int nt = 0; nt < NT; ++nt) {
        v2f b;
        b.x = Bp[(size_t)k0 * N + nt * 16];          // B[k0+kh  , n]
        b.y = Bp[(size_t)(k0 + 1) * N + nt * 16];    // B[k0+kh+1, n]
        acc[nt] = wmma_f32_k4(a, b, acc[nt]);
      }
    }
  }

  const int chi = (lane >> 4) << 3;                  // 0 or 8
  #pragma unroll
  for (int nt = 0; nt < NT; ++nt) {
    #pragma unroll
    for (int j = 0; j < 8; ++j) {
      float v = acc[nt][j];
      if (do_relu) v = fmaxf(v, 0.0f);
      C[(size_t)(m0 + j + chi) * N + n0 + nt * 16 + (lane & 15)] = v;
    }
  }
}

// ---------------------------------------------------------------------------
// Gate: G[tok,n] = logsigmoid(Tmp[tok,:16] @ Wg2[:,n] + bg2[n]) / 16
// ---------------------------------------------------------------------------
__global__ void __launch_bounds__(256)
gate_kernel(const float* __restrict__ Tmp, const float* __restrict__ Wg2,
            const float* __restrict__ bg2, float* __restrict__ G, int M)
{
  int idx = blockIdx.x * blockDim.x + threadIdx.x;
  if (idx >= M * GLA_KVW) return;
  int tok = idx >> 8;
  int n   = idx & 255;
  const float* tp = Tmp + (size_t)tok * 16;
  float acc = bg2[n];
  #pragma unroll
  for (int j = 0; j < 16; ++j) acc += tp[j] * Wg2[j * GLA_KVW + n];
  // stable logsigmoid(x) = min(x,0) - log1p(exp(-|x|))
  float ls = fminf(acc, 0.0f) - log1pf(__expf(-fabsf(acc)));
  G[idx] = ls * (1.0f / 16.0f);
}

// ---------------------------------------------------------------------------
// Tensor Data Mover: DMA a 2D tile (tile_w x tile_h elements, f32, row stride
// stride_elems) from global memory into LDS at byte offset lds_off.
// D# built per ISA 8.3/8.4: group0 = {count=1, lds_addr, global_addr, type=2},
// group1 = {data_size=4B, tensor dims, tile dims, 48-bit dim0 stride}.
// Issued per-wave; completion via TENSORcnt.
// ---------------------------------------------------------------------------
__device__ __forceinline__ void tdm_load_2d(const float* gptr, unsigned lds_off,
                                            unsigned tile_w, unsigned tile_h,
                                            unsigned long long stride_elems)
{
  const unsigned long long ga = (unsigned long long)(uintptr_t)gptr;
  u32x4 g0;
  g0.x = 1u;                                          // count=1, user descriptor
  g0.y = lds_off;                                     // lds_addr (bytes)
  g0.z = (unsigned)(ga & 0xffffffffu);                // global_addr[31:0]
  g0.w = (unsigned)((ga >> 32) & 0x01ffffffu)         // global_addr[56:32]
         | (2u << 30);                                // type = 2 ("image")
  u32x8 g1;
  g1.s0 = (2u << 16);                                 // data_size = 2 -> 4 bytes
  g1.s1 = (tile_w & 0xffffu) << 16;                   // tensor_dim0[15:0]
  g1.s2 = ((tile_w >> 16) & 0xffffu)                  // tensor_dim0[31:16]
          | ((tile_h & 0xffffu) << 16);               // tensor_dim1[15:0]
  g1.s3 = ((tile_h >> 16) & 0xffffu)                  // tensor_dim1[31:16]
          | ((tile_w & 0xffffu) << 16);               // tile_dim0
  g1.s4 = (tile_h & 0xffffu);                         // tile_dim1 (tile_dim2=0)
  g1.s5 = (unsigned)(stride_elems & 0xffffffffu);     // tensor_dim0_stride[31:0]
  g1.s6 = (unsigned)((stride_elems >> 32) & 0xffffu); // stride[47:32]
  g1.s7 = 0u;
  asm volatile("tensor_load_to_lds %0, %1" :: "s"(g0), "s"(g1) : "memory");
}

__device__ __forceinline__ unsigned lds_byte_off(const void* p) {
  // generic LDS pointer = {shared_base, 32'offset}; low 32 bits are the offset
  return (unsigned)(uintptr_t)p;
}

// ---------------------------------------------------------------------------
// GLA recurrence. One block per (batch, kv-head); 4 q-head subgroups of 64
// threads; each thread owns a full state column S[0..63][c] in VGPRs, so the
// output o[c] = sum_r q[r]*S[r][c] needs no cross-lane reduction.
// k/v/g/q are DMA'd into LDS by the TDM in 32-timestep chunks; g is turned
// into exp(g) in-place; the q scale D^-0.5 is folded into the final output.
// ---------------------------------------------------------------------------
#define SCAN_CH 32

__global__ void __launch_bounds__(256)
gla_scan(const float* __restrict__ Q, const float* __restrict__ Kb,
         const float* __restrict__ Vb, const float* __restrict__ Gb,
         float* __restrict__ O)
{
  __shared__ float s_k[SCAN_CH][GLA_D];
  __shared__ float s_v[SCAN_CH][GLA_D];
  __shared__ float s_g[SCAN_CH][GLA_D];        // raw g, converted to exp(g)
  __shared__ float s_q[SCAN_CH][4 * GLA_D];

  const int b   = blockIdx.x >> 2;     // / HKV
  const int kvh = blockIdx.x & 3;      // % HKV
  const int tid = threadIdx.x;
  const int sub = tid >> 6;            // q head within GQA group (0..3)
  const int c   = tid & 63;            // state/value column
  const int qh  = kvh * 4 + sub;       // repeat_interleave mapping

  float S[GLA_D];
  #pragma unroll
  for (int r = 0; r < GLA_D; ++r) S[r] = 0.0f;

  const size_t tok0 = (size_t)b * GLA_T;
  const unsigned off_k = lds_byte_off(&s_k[0][0]);
  const unsigned off_v = lds_byte_off(&s_v[0][0]);
  const unsigned off_g = lds_byte_off(&s_g[0][0]);
  const unsigned off_q = lds_byte_off(&s_q[0][0]);

  for (int t0 = 0; t0 < GLA_T; t0 += SCAN_CH) {
    const size_t r0 = tok0 + t0;
    if (tid < 32) {  // wave 0 drives the TDM
      tdm_load_2d(Kb + r0 * GLA_KVW + kvh * GLA_D, off_k, GLA_D, SCAN_CH, GLA_KVW);
      tdm_load_2d(Vb + r0 * GLA_KVW + kvh * GLA_D, off_v, GLA_D, SCAN_CH, GLA_KVW);
      tdm_load_2d(Gb + r0 * GLA_KVW + kvh * GLA_D, off_g, GLA_D, SCAN_CH, GLA_KVW);
      tdm_load_2d(Q  + r0 * GLA_HID + kvh * 256,   off_q, 4 * GLA_D, SCAN_CH, GLA_HID);
      __builtin_amdgcn_s_wait_tensorcnt(0);
    }
    __syncthreads();

    // g -> exp(g) in place (2048 elems, 8 per thread)
    {
      float* gf = &s_g[0][0];
      #pragma unroll
      for (int i = 0; i < (SCAN_CH * GLA_D) / 256; ++i)
        gf[tid + i * 256] = __expf(gf[tid + i * 256]);
    }
    __syncthreads();

    for (int tt = 0; tt < SCAN_CH; ++tt) {
      const float vv = s_v[tt][c];
      const float* qp = &s_q[tt][sub * GLA_D];
      float o = 0.0f;
      #pragma unroll
      for (int r = 0; r < GLA_D; ++r) {
        S[r] = s_g[tt][r] * S[r] + s_k[tt][r] * vv;   // LDS broadcast reads
        o += qp[r] * S[r];
      }
      O[(r0 + tt) * GLA_HID + qh * GLA_D + c] = o * 0.125f;  // D^-0.5
    }
    __syncthreads();
  }
}

// ---------------------------------------------------------------------------
// Per-(token,head) RMSNorm over D=64. One wave32 per row: lane holds elements
// {lane, lane+32}; sum-of-squares reduced with wave32 shuffles.
// ---------------------------------------------------------------------------
__global__ void __launch_bounds__(256)
rmsnorm_kernel(float* __restrict__ O, const float* __restrict__ w, int rows)
{
  const int lane = threadIdx.x & 31;
  const int row  = blockIdx.x * 8 + (threadIdx.x >> 5);
  if (row >= rows) return;
  float* p = O + (size_t)row * GLA_D;
  float a  = p[lane];
  float bv = p[lane + 32];
  float ss = a * a + bv * bv;
  #pragma unroll
  for (int m = 16; m > 0; m >>= 1) ss += __shfl_xor(ss, m, 32);
  float r = rsqrtf(ss * (1.0f / GLA_D) + 1e-6f);
  p[lane]      = a  * r * w[lane];
  p[lane + 32] = bv * r * w[lane + 32];
}

// ---------------------------------------------------------------------------
extern "C" void kernel_launch(void* const* d_in, const int* in_sizes, int n_in,
                              void* d_out, int out_size, void* d_ws, size_t ws_size,
                              hipStream_t stream) {
  const float* h    = (const float*)d_in[0];   // [B,T,HID]
  const float* Wq   = (const float*)d_in[1];   // [HID, H*D]
  const float* Wk   = (const float*)d_in[2];   // [HID, HKV*D]
  const float* Wv   = (const float*)d_in[3];   // [HID, HKV*D]
  const float* Wo   = (const float*)d_in[4];   // [H*D, HID]
  const float* Wg1  = (const float*)d_in[5];   // [HID, 16]
  const float* Wg2  = (const float*)d_in[6];   // [16, HKV*D]
  const float* bg2  = (const float*)d_in[7];   // [HKV*D]
  const float* gnw  = (const float*)d_in[8];   // [D]
  float* out = (float*)d_out;

  // workspace layout (fp32)
  float* ws  = (float*)d_ws;
  float* Qb  = ws;                                   // 4096*1024
  float* Kb  = Qb  + (size_t)GLA_TOK * GLA_HID;      // 4096*256
  float* Vb  = Kb  + (size_t)GLA_TOK * GLA_KVW;      // 4096*256
  float* Gb  = Vb  + (size_t)GLA_TOK * GLA_KVW;      // 4096*256
  float* Tmp = Gb  + (size_t)GLA_TOK * GLA_KVW;      // 4096*16
  float* Ob  = Tmp + (size_t)GLA_TOK * 16;           // 4096*1024

  auto gemm_blocks = [](int M, int N, int nt) {
    int tiles = (M >> 4) * (N / (16 * nt));
    return (tiles + 7) / 8;
  };

  // 1) projections (WMMA f32)
  gemm_f32_wmma<4><<<gemm_blocks(GLA_TOK, GLA_HID, 4), 256, 0, stream>>>(
      h, Wq, Qb, GLA_TOK, GLA_HID, GLA_HID, /*relu=*/1);
  gemm_f32_wmma<4><<<gemm_blocks(GLA_TOK, GLA_KVW, 4), 256, 0, stream>>>(
      h, Wk, Kb, GLA_TOK, GLA_KVW, GLA_HID, /*relu=*/1);
  gemm_f32_wmma<4><<<gemm_blocks(GLA_TOK, GLA_KVW, 4), 256, 0, stream>>>(
      h, Wv, Vb, GLA_TOK, GLA_KVW, GLA_HID, /*relu=*/0);
  gemm_f32_wmma<1><<<gemm_blocks(GLA_TOK, 16, 1), 256, 0, stream>>>(
      h, Wg1, Tmp, GLA_TOK, 16, GLA_HID, /*relu=*/0);

  // 2) low-rank gate -> log-space decay
  {
    int n = GLA_TOK * GLA_KVW;
    gate_kernel<<<(n + 255) / 256, 256, 0, stream>>>(Tmp, Wg2, bg2, Gb, GLA_TOK);
  }

  // 3) recurrence: one block per (batch, kv-head), TDM-staged LDS
  gla_scan<<<GLA_B * GLA_HKV, 256, 0, stream>>>(Qb, Kb, Vb, Gb, Ob);

  // 4) per-head RMSNorm (rows = B*T*H)
  {
    int rows = GLA_TOK * GLA_H;
    rmsnorm_kernel<<<(rows + 7) / 8, 256, 0, stream>>>(Ob, gnw, rows);
  }

  // 5) output projection (WMMA f32)
  gemm_f32_wmma<4><<<gemm_blocks(GLA_TOK, GLA_HID, 4), 256, 0, stream>>>(
      Ob, Wo, out, GLA_TOK, GLA_HID, GLA_HID, /*relu=*/0);
}